// CRFLayer_48249662603501
// MI455X (gfx1250) — compile-verified
//
#include <hip/hip_runtime.h>
#include <hip/hip_bf16.h>

#define Hh    2048
#define Ww    2048
#define NC    4
#define RAD   9
#define KLEN  19
#define SIG2  9.0f     // sigma^2, sigma = 3
#define NITER 5
#define NT    4        // sub-tiles along the convolution axis (64-long strip)
#define SLAB  84       // 16*NT + 18 taps halo + 2 pad (s indices 0..83)

typedef __attribute__((ext_vector_type(2))) float v2f;
typedef __attribute__((ext_vector_type(8))) float v8f;

// Fill zero-padded tap table: sKpad[16+j] = gauss tap j for j in [0,18], else 0.
// Call with tid; single phase, no intra-phase races.
__device__ __forceinline__ void fill_taps(float* sKpad, int tid) {
  if (tid < 64) {
    float v = 0.0f;
    if (tid >= 16 && tid < 16 + KLEN) {
      float x = (float)(tid - 16 - RAD);
      v = __expf(-0.5f * x * x / SIG2);
    }
    sKpad[tid] = v;
  }
}

// sum of taps that land in-bounds => G(ones) along one axis (separable norm)
__device__ __forceinline__ float border_sum(const float* sK16, int pos, int n) {
  float s = 0.0f;
#pragma unroll
  for (int t = 0; t < KLEN; ++t) {
    int p = pos + t - RAD;
    s += (p >= 0 && p < n) ? sK16[t] : 0.0f;
  }
  return s;
}

// Per-lane A fragment of the 16x36 Gaussian Toeplitz, 9 K-blocks of 4.
// ISA A-layout (16x4 f32): lanes 0-15 -> M=lane,   K=0 (v0) / K=1 (v1);
//                          lanes 16-31 -> M=lane-16, K=2 (v0) / K=3 (v1).
// Branchless via the zero-padded tap table (index range [-15..36] + 16).
__device__ __forceinline__ void build_afrag(v2f* a, const float* sK16, int lane) {
  const int m = lane & 15;
  const int khalf = (lane < 16) ? 0 : 2;
#pragma unroll
  for (int kb = 0; kb < 9; ++kb) {
    int s0 = 4 * kb + khalf;
    a[kb].x = sK16[s0 - m];
    a[kb].y = sK16[s0 + 1 - m];
  }
}

// ---------------------------------------------------------------------------
// Kernel 0: Q = softmax(-unary), interleaved [H,W,4], float4 in / float4 out.
// ---------------------------------------------------------------------------
__global__ __launch_bounds__(256) void crf_init_softmax(
    const float* __restrict__ unary, float* __restrict__ Q) {
  int idx = blockIdx.x * blockDim.x + threadIdx.x;
  if (idx >= Hh * Ww) return;
  float4 u = ((const float4*)unary)[idx];
  float z0 = -u.x, z1 = -u.y, z2 = -u.z, z3 = -u.w;
  float mx = fmaxf(fmaxf(z0, z1), fmaxf(z2, z3));
  float e0 = __expf(z0 - mx), e1 = __expf(z1 - mx);
  float e2 = __expf(z2 - mx), e3 = __expf(z3 - mx);
  float inv = 1.0f / (e0 + e1 + e2 + e3);
  ((float4*)Q)[idx] = make_float4(e0 * inv, e1 * inv, e2 * inv, e3 * inv);
}

// ---------------------------------------------------------------------------
// Pass A: vertical blur of (Q * norm) via WMMA Toeplitz; tmp is planar [C][H][W].
// Block = 128 threads = 4 waves (one channel each); strip = 64 rows x 16 cols
// (4 sub-tiles along y -> 36 WMMAs/wave, halo amortized 4x).
// ---------------------------------------------------------------------------
__global__ __launch_bounds__(128) void crf_blur_v(
    const float* __restrict__ Q, float* __restrict__ tmp) {
  __shared__ float sP[NC][SLAB][16];   // channel-major: conflict-free ds reads
  __shared__ float sKpad[64];
  __shared__ float sSy[SLAB];
  __shared__ float sSx[16];

  const int tid = threadIdx.x;
  const int xbase = blockIdx.x * 16;
  const int ybase = blockIdx.y * 64;

  fill_taps(sKpad, tid);
  __syncthreads();
  const float* sK16 = &sKpad[16];

  if (tid < SLAB)                       sSy[tid]      = border_sum(sK16, ybase + tid - RAD, Hh);
  else if (tid >= 96 && tid < 96 + 16)  sSx[tid - 96] = border_sum(sK16, xbase + tid - 96, Ww);
  __syncthreads();

  // Stage SLAB rows x 16 cols of P = Q * norm, zero-padded vertically.
  for (int p = tid; p < SLAB * 16; p += 128) {
    int s = p >> 4, n = p & 15;
    int gy = ybase + s - RAD;
    float4 q = make_float4(0.f, 0.f, 0.f, 0.f);
    float nv = 0.0f;
    if (gy >= 0 && gy < Hh) {
      q = ((const float4*)Q)[(size_t)gy * Ww + (xbase + n)];
      nv = 1.0f / (sqrtf(sSy[s] * sSx[n]) + 1e-20f);
    }
    sP[0][s][n] = q.x * nv;
    sP[1][s][n] = q.y * nv;
    sP[2][s][n] = q.z * nv;
    sP[3][s][n] = q.w * nv;
  }
  __syncthreads();

  const int wave = tid >> 5;      // channel
  const int lane = tid & 31;
  const int n = lane & 15;
  const int khalf = (lane < 16) ? 0 : 2;
  const int rbase = (lane < 16) ? 0 : 8;

  v2f afrag[9];
  build_afrag(afrag, sK16, lane);

#pragma unroll
  for (int t = 0; t < NT; ++t) {
    v8f acc = {};
#pragma unroll
    for (int kb = 0; kb < 9; ++kb) {
      int s0 = 16 * t + 4 * kb + khalf;
      v2f b;                      // B 4x16: lanes 0-15 K=0,1; lanes 16-31 K=2,3
      b.x = sP[wave][s0][n];
      b.y = sP[wave][s0 + 1][n];
      acc = __builtin_amdgcn_wmma_f32_16x16x4_f32(
          false, afrag[kb], false, b, (short)0, acc, false, false);
    }
    // D layout: VGPR v -> row M=v (lanes 0-15) / M=v+8 (lanes 16-31), col = n.
    float* out = tmp + (size_t)wave * Hh * Ww
                     + (size_t)(ybase + 16 * t + rbase) * Ww + (xbase + n);
#pragma unroll
    for (int v = 0; v < 8; ++v) out[(size_t)v * Ww] = acc[v];
  }
}

// ---------------------------------------------------------------------------
// Pass B: horizontal blur of tmp via WMMA Toeplitz, * norm, 4x4 channel mix
// with M = SW @ CM, fused softmax(-unary - pairwise) -> Q (interleaved).
// Strip = 16 rows x 64 cols (4 sub-tiles along x).
// ---------------------------------------------------------------------------
__global__ __launch_bounds__(128) void crf_blur_h_update(
    const float* __restrict__ tmp, const float* __restrict__ unary,
    const float* __restrict__ SW, const float* __restrict__ CM,
    float* __restrict__ Q) {
  __shared__ float sT[NC][SLAB][17];   // [c][x-tap s][y], padded: conflict-free
  __shared__ float sSp[NC][64][18];    // sp [c][x_local][y_local], padded
  __shared__ float sKpad[64];
  __shared__ float sSy[16];
  __shared__ float sSx[64];

  const int tid = threadIdx.x;
  const int xbase = blockIdx.x * 64;
  const int ybase = blockIdx.y * 16;

  fill_taps(sKpad, tid);
  __syncthreads();
  const float* sK16 = &sKpad[16];

  if (tid < 16)                    sSy[tid]      = border_sum(sK16, ybase + tid, Hh);
  else if (tid >= 32 && tid < 96)  sSx[tid - 32] = border_sum(sK16, xbase + tid - 32, Ww);

  // Stage transposed slab: sT[c][s][n] = tmp[c][ybase+n][xbase+s-9], zero-pad x.
  // e -> s fastest so global reads are row-contiguous (336B runs).
  __syncthreads();
  for (int e = tid; e < NC * 16 * SLAB; e += 128) {
    int c = e / (16 * SLAB);
    int rem = e - c * (16 * SLAB);
    int n = rem / SLAB;
    int s = rem - n * SLAB;
    int gx = xbase + s - RAD;
    float v = 0.0f;
    if (gx >= 0 && gx < Ww)
      v = tmp[(size_t)c * Hh * Ww + (size_t)(ybase + n) * Ww + gx];
    sT[c][s][n] = v;
  }
  __syncthreads();

  const int wave = tid >> 5;
  const int lane = tid & 31;
  const int n = lane & 15;           // y_local for B cols / D cols
  const int khalf = (lane < 16) ? 0 : 2;
  const int rbase = (lane < 16) ? 0 : 8;

  v2f afrag[9];
  build_afrag(afrag, sK16, lane);

#pragma unroll
  for (int t = 0; t < NT; ++t) {
    v8f acc = {};
#pragma unroll
    for (int kb = 0; kb < 9; ++kb) {
      int s0 = 16 * t + 4 * kb + khalf;
      v2f b;
      b.x = sT[wave][s0][n];
      b.y = sT[wave][s0 + 1][n];
      acc = __builtin_amdgcn_wmma_f32_16x16x4_f32(
          false, afrag[kb], false, b, (short)0, acc, false, false);
    }
    // sp = D * norm; D rows = x_local (within sub-tile), cols = y_local.
#pragma unroll
    for (int v = 0; v < 8; ++v) {
      int xl = 16 * t + rbase + v;
      float nv = 1.0f / (sqrtf(sSy[n] * sSx[xl]) + 1e-20f);
      sSp[wave][xl][n] = acc[v] * nv;
    }
  }
  __syncthreads();

  // Combined mix matrix M = SW @ CM (4x4, broadcast scalar loads).
  float M4[4][4];
#pragma unroll
  for (int i = 0; i < 4; ++i)
#pragma unroll
    for (int j = 0; j < 4; ++j) {
      float a = 0.0f;
#pragma unroll
      for (int k = 0; k < 4; ++k) a += SW[i * 4 + k] * CM[k * 4 + j];
      M4[i][j] = a;
    }

  // Per-pixel: pairwise = sp @ M, Q = softmax(-unary - pairwise).
#pragma unroll
  for (int pp = 0; pp < 8; ++pp) {
    int p = tid + pp * 128;            // 16 x 64 = 1024 pixels
    int yl = p >> 6, xl = p & 63;
    float s0 = sSp[0][xl][yl], s1 = sSp[1][xl][yl];
    float s2 = sSp[2][xl][yl], s3 = sSp[3][xl][yl];
    float pw0 = s0 * M4[0][0] + s1 * M4[1][0] + s2 * M4[2][0] + s3 * M4[3][0];
    float pw1 = s0 * M4[0][1] + s1 * M4[1][1] + s2 * M4[2][1] + s3 * M4[3][1];
    float pw2 = s0 * M4[0][2] + s1 * M4[1][2] + s2 * M4[2][2] + s3 * M4[3][2];
    float pw3 = s0 * M4[0][3] + s1 * M4[1][3] + s2 * M4[2][3] + s3 * M4[3][3];
    size_t gidx = (size_t)(ybase + yl) * Ww + (xbase + xl);
    float4 u = ((const float4*)unary)[gidx];
    float z0 = -u.x - pw0, z1 = -u.y - pw1, z2 = -u.z - pw2, z3 = -u.w - pw3;
    float mx = fmaxf(fmaxf(z0, z1), fmaxf(z2, z3));
    float e0 = __expf(z0 - mx), e1 = __expf(z1 - mx);
    float e2 = __expf(z2 - mx), e3 = __expf(z3 - mx);
    float inv = 1.0f / (e0 + e1 + e2 + e3);
    ((float4*)Q)[gidx] = make_float4(e0 * inv, e1 * inv, e2 * inv, e3 * inv);
  }
}

// ---------------------------------------------------------------------------
extern "C" void kernel_launch(void* const* d_in, const int* in_sizes, int n_in,
                              void* d_out, int out_size, void* d_ws, size_t ws_size,
                              hipStream_t stream) {
  const float* unary = (const float*)d_in[0];
  // d_in[1] (image) is unused: truncated reference applies only spatial kernel.
  const float* SW = (const float*)d_in[2];
  const float* CM = (const float*)d_in[3];
  float* Q   = (float*)d_out;            // Q lives in d_out across iterations
  float* tmp = (float*)d_ws;             // planar [C][H][W] vertical-blur scratch

  const int npix = Hh * Ww;
  crf_init_softmax<<<(npix + 255) / 256, 256, 0, stream>>>(unary, Q);

  dim3 gridA(Ww / 16, Hh / 64);          // 128 x 32
  dim3 gridB(Ww / 64, Hh / 16);          // 32 x 128
  for (int it = 0; it < NITER; ++it) {
    crf_blur_v<<<gridA, 128, 0, stream>>>(Q, tmp);
    crf_blur_h_update<<<gridB, 128, 0, stream>>>(tmp, unary, SW, CM, Q);
  }
}